// SparseGraphTransformerLayer_33749853012145
// MI455X (gfx1250) — compile-verified
//
#include <hip/hip_runtime.h>
#include <hip/hip_bf16.h>
#include <float.h>

// ---------------------------------------------------------------------------
// SparseGraphTransformerLayer for MI455X (gfx1250, wave32)
// GEMMs use V_WMMA_F32_16X16X4_F32 (exact fp32 matrix path on CDNA5).
// 64-row M-tiles: each B fragment feeds 4 WMMAs (4x register reuse).
// ---------------------------------------------------------------------------

typedef float v2f __attribute__((ext_vector_type(2)));
typedef float v8f __attribute__((ext_vector_type(8)));

__device__ __forceinline__ unsigned fenc(float f) {
    unsigned u = __float_as_uint(f);
    return (u >> 31) ? ~u : (u | 0x80000000u);
}
__device__ __forceinline__ float fdec(unsigned u) {
    return (u >> 31) ? __uint_as_float(u & 0x7FFFFFFFu) : __uint_as_float(~u);
}

// ---------------- LayerNorm: one wave32 per 128-wide row --------------------
__global__ __launch_bounds__(256) void ln128_kernel(
    const float* __restrict__ x, const float* __restrict__ g,
    const float* __restrict__ b, float* __restrict__ y, int n)
{
    int row = blockIdx.x * 8 + (threadIdx.x >> 5);
    if (row >= n) return;
    int lane = threadIdx.x & 31;
    const float4* xr = reinterpret_cast<const float4*>(x + (size_t)row * 128);
    float4 v = xr[lane];
    float s = v.x + v.y + v.z + v.w;
    #pragma unroll
    for (int off = 16; off; off >>= 1) s += __shfl_xor(s, off, 32);
    float mu = s * (1.0f / 128.0f);
    float dx0 = v.x - mu, dx1 = v.y - mu, dx2 = v.z - mu, dx3 = v.w - mu;
    float q = dx0 * dx0 + dx1 * dx1 + dx2 * dx2 + dx3 * dx3;
    #pragma unroll
    for (int off = 16; off; off >>= 1) q += __shfl_xor(q, off, 32);
    float inv = rsqrtf(q * (1.0f / 128.0f) + 1e-5f);
    const float4 gg = reinterpret_cast<const float4*>(g)[lane];
    const float4 bb = reinterpret_cast<const float4*>(b)[lane];
    float4 o;
    o.x = dx0 * inv * gg.x + bb.x;
    o.y = dx1 * inv * gg.y + bb.y;
    o.z = dx2 * inv * gg.z + bb.z;
    o.w = dx3 * inv * gg.w + bb.w;
    reinterpret_cast<float4*>(y + (size_t)row * 128)[lane] = o;
}

// ---------------- WMMA fp32 GEMM: D = act(A@W + bias) + resid ---------------
// Block: 256 threads (8 waves). Each block: 64 rows x 128 cols of D.
// grid.x = ceil(n/64), grid.y = P/128. K blocked in 128-chunks via 32KB LDS.
// Wave w owns columns [col0+16w, col0+16w+16): one B frag -> 4 WMMAs.
// zdiv (optional): divide A row-block by softmax partition sums while staging
//   (A must be [n x 128] laid out as 4 heads x 32; empty segment -> 0).
// Frag layouts (V_WMMA_F32_16X16X4_F32):
//   A: lanes 0-15 k=kk,kk+1 ; lanes 16-31 k=kk+2,kk+3 (row = lane%16)
//   B: VGPR0 = row kk+2*half, VGPR1 = row kk+1+2*half (col = lane%16)
//   C/D: acc[j] = row j+8*half (col = lane%16)
#define KC 128
__global__ __launch_bounds__(256) void gemm_wmma_f32_kernel(
    const float* __restrict__ A, const float* __restrict__ W,
    const float* __restrict__ bias, const float* __restrict__ resid,
    const float* __restrict__ zdiv, float* __restrict__ D,
    int n, int K, int P, int act /*0 none, 1 gelu*/)
{
    __shared__ float lds[64 * KC];
    const int row0 = blockIdx.x * 64;
    const int col0 = blockIdx.y * 128;
    const int tid  = threadIdx.x;
    const int wave = tid >> 5;
    const int lane = tid & 31;
    const int half = lane >> 4;
    const int l16  = lane & 15;
    const int ncol = col0 + wave * 16 + l16;

    v8f acc0 = {}, acc1 = {}, acc2 = {}, acc3 = {};
    const float* wp = W + (size_t)(half * 2) * P + ncol;   // steps by 4*P per k-step

    for (int kb = 0; kb < K; kb += KC) {
        // ---- stage A[row0..row0+63][kb..kb+KC) into LDS as float4 ----
        for (int i = tid; i < 64 * (KC / 4); i += 256) {
            int r = i >> 5;              // KC/4 == 32
            int c = i & 31;
            int gr = row0 + r;
            float4 v = make_float4(0.f, 0.f, 0.f, 0.f);
            if (gr < n) {
                v = reinterpret_cast<const float4*>(A + (size_t)gr * K + kb)[c];
                if (zdiv) {
                    float zz = zdiv[gr * 4 + (((kb + 4 * c) >> 5) & 3)];
                    float inv = (zz > 0.0f) ? 1.0f / zz : 0.0f;
                    v.x *= inv; v.y *= inv; v.z *= inv; v.w *= inv;
                }
            }
            reinterpret_cast<float4*>(lds)[i] = v;
        }
        __syncthreads();
        // ---- 32 k-steps, 4 M-tiles each: 128 WMMAs per chunk per wave ----
        #pragma unroll 4
        for (int kk = 0; kk < KC; kk += 4) {
            const int ka = kk + half * 2;
            v2f b; b.x = wp[0]; b.y = wp[P];
            v2f a0, a1, a2, a3;
            a0.x = lds[(l16     ) * KC + ka]; a0.y = lds[(l16     ) * KC + ka + 1];
            a1.x = lds[(l16 + 16) * KC + ka]; a1.y = lds[(l16 + 16) * KC + ka + 1];
            a2.x = lds[(l16 + 32) * KC + ka]; a2.y = lds[(l16 + 32) * KC + ka + 1];
            a3.x = lds[(l16 + 48) * KC + ka]; a3.y = lds[(l16 + 48) * KC + ka + 1];
            acc0 = __builtin_amdgcn_wmma_f32_16x16x4_f32(false, a0, false, b, (short)0, acc0, false, false);
            acc1 = __builtin_amdgcn_wmma_f32_16x16x4_f32(false, a1, false, b, (short)0, acc1, false, false);
            acc2 = __builtin_amdgcn_wmma_f32_16x16x4_f32(false, a2, false, b, (short)0, acc2, false, false);
            acc3 = __builtin_amdgcn_wmma_f32_16x16x4_f32(false, a3, false, b, (short)0, acc3, false, false);
            wp += (size_t)4 * P;
        }
        __syncthreads();
    }

    const float bv = bias ? bias[ncol] : 0.0f;
    v8f accs[4] = {acc0, acc1, acc2, acc3};
    #pragma unroll
    for (int t = 0; t < 4; ++t) {
        #pragma unroll
        for (int j = 0; j < 8; ++j) {
            int r = row0 + t * 16 + j + 8 * half;
            if (r >= n) continue;
            float v = accs[t][j] + bv;
            if (act == 1) v = 0.5f * v * (1.0f + erff(v * 0.70710678118654752f));
            if (resid) v += resid[(size_t)r * P + ncol];
            D[(size_t)r * P + ncol] = v;
        }
    }
}

// ---------------- init: zero z/agg, set segment-max identity ----------------
__global__ __launch_bounds__(256) void attn_init_kernel(
    float* __restrict__ agg, float* __restrict__ z,
    unsigned* __restrict__ menc, int n)
{
    int i = blockIdx.x * blockDim.x + threadIdx.x;
    if (i < n * 128) agg[i] = 0.0f;
    if (i < n * 4) { z[i] = 0.0f; menc[i] = fenc(-FLT_MAX); }
}

// ---------------- per-edge scores + segment max -----------------------------
__global__ __launch_bounds__(256) void edge_score_kernel(
    const int* __restrict__ ei, const int* __restrict__ degree,
    const float* __restrict__ Q, const float* __restrict__ Km,
    const float* __restrict__ degs, const float* __restrict__ degd,
    float* __restrict__ alpha, unsigned* __restrict__ menc, int E_)
{
    int t = blockIdx.x * blockDim.x + threadIdx.x;
    if (t >= E_ * 4) return;
    int e = t >> 2, h = t & 3;
    int src = ei[e], dst = ei[E_ + e];
    const float4* q = reinterpret_cast<const float4*>(Q  + (size_t)dst * 128 + h * 32);
    const float4* k = reinterpret_cast<const float4*>(Km + (size_t)src * 128 + h * 32);
    float s = 0.0f;
    #pragma unroll
    for (int i = 0; i < 8; ++i) {
        float4 a = q[i], b = k[i];
        s += a.x * b.x + a.y * b.y + a.z * b.z + a.w * b.w;
    }
    s *= 0.17677669529663689f;                  // 1/sqrt(32)
    int ds = degree[src]; ds = ds > 511 ? 511 : ds;
    int dd = degree[dst]; dd = dd > 511 ? 511 : dd;
    s += degs[ds * 4 + h] + degd[dd * 4 + h];
    alpha[t] = s;
    atomicMax(&menc[dst * 4 + h], fenc(s));
}

// ---------------- exp + unnormalized weighted scatter-add -------------------
__global__ __launch_bounds__(256) void edge_agg_kernel(
    const int* __restrict__ ei, const float* __restrict__ V,
    const float* __restrict__ alpha, const unsigned* __restrict__ menc,
    float* __restrict__ z, float* __restrict__ agg, int E_)
{
    int t = blockIdx.x * blockDim.x + threadIdx.x;
    if (t >= E_ * 4) return;
    int e = t >> 2, h = t & 3;
    int src = ei[e], dst = ei[E_ + e];
    float w = expf(alpha[t] - fdec(menc[dst * 4 + h]));
    atomicAdd(&z[dst * 4 + h], w);
    const float* v = V + (size_t)src * 128 + h * 32;
    float* o = agg + (size_t)dst * 128 + h * 32;
    #pragma unroll
    for (int c = 0; c < 32; ++c) atomicAdd(&o[c], w * v[c]);
}

// ---------------------------------------------------------------------------
extern "C" void kernel_launch(void* const* d_in, const int* in_sizes, int n_in,
                              void* d_out, int out_size, void* d_ws, size_t ws_size,
                              hipStream_t stream)
{
    const float* x    = (const float*)d_in[0];
    const int*   ei   = (const int*)  d_in[1];
    const int*   deg  = (const int*)  d_in[2];
    const float* Wq   = (const float*)d_in[3];
    const float* Wk   = (const float*)d_in[4];
    const float* Wv   = (const float*)d_in[5];
    const float* Wo   = (const float*)d_in[6];
    const float* bo   = (const float*)d_in[7];
    const float* degs = (const float*)d_in[8];
    const float* degd = (const float*)d_in[9];
    const float* g1   = (const float*)d_in[10];
    const float* b1   = (const float*)d_in[11];
    const float* g2   = (const float*)d_in[12];
    const float* b2   = (const float*)d_in[13];
    const float* Wf1  = (const float*)d_in[14];
    const float* bf1  = (const float*)d_in[15];
    const float* Wf2  = (const float*)d_in[16];
    const float* bf2  = (const float*)d_in[17];

    const int N_ = in_sizes[0] / 128;   // 50000
    const int E_ = in_sizes[1] / 2;     // 800000

    float* ws    = (float*)d_ws;
    float* xn    = ws;                                  // N*128 (reused as h1)
    float* Q     = ws + (size_t)1 * N_ * 128;
    float* Km    = ws + (size_t)2 * N_ * 128;
    float* V     = ws + (size_t)3 * N_ * 128;
    float* agg   = ws + (size_t)4 * N_ * 128;
    float* ff    = ws + (size_t)5 * N_ * 128;           // N*512
    float* alpha = ws + (size_t)9 * N_ * 128;           // E*4
    float* z     = alpha + (size_t)E_ * 4;              // N*4
    unsigned* menc = (unsigned*)(z + (size_t)N_ * 4);   // N*4
    float* out   = (float*)d_out;

    const dim3 blk(256);
    const dim3 gLN((N_ + 7) / 8);
    const dim3 gM128((N_ + 63) / 64, 1);
    const dim3 gM512((N_ + 63) / 64, 4);
    const int  edgeBlocks = (E_ * 4 + 255) / 256;
    const int  nodeBlocks = (N_ * 128 + 255) / 256;

    // 1) x_norm = LN(x)
    ln128_kernel<<<gLN, blk, 0, stream>>>(x, g1, b1, xn, N_);
    // 2) Q,K,V projections (WMMA fp32)
    gemm_wmma_f32_kernel<<<gM128, blk, 0, stream>>>(xn, Wq, nullptr, nullptr, nullptr, Q,  N_, 128, 128, 0);
    gemm_wmma_f32_kernel<<<gM128, blk, 0, stream>>>(xn, Wk, nullptr, nullptr, nullptr, Km, N_, 128, 128, 0);
    gemm_wmma_f32_kernel<<<gM128, blk, 0, stream>>>(xn, Wv, nullptr, nullptr, nullptr, V,  N_, 128, 128, 0);
    // 3) attention state init + edge scores + segment softmax aggregation
    attn_init_kernel<<<nodeBlocks, blk, 0, stream>>>(agg, z, menc, N_);
    edge_score_kernel<<<edgeBlocks, blk, 0, stream>>>(ei, deg, Q, Km, degs, degd, alpha, menc, E_);
    edge_agg_kernel<<<edgeBlocks, blk, 0, stream>>>(ei, V, alpha, menc, z, agg, E_);
    // 4) out = (agg/z)@Wo + bo + x   (softmax normalization fused into staging)
    gemm_wmma_f32_kernel<<<gM128, blk, 0, stream>>>(agg, Wo, bo, x, z, out, N_, 128, 128, 0);
    // 5) h1 = LN(out); FFN with exact GELU; final = out + h
    ln128_kernel<<<gLN, blk, 0, stream>>>(out, g2, b2, xn, N_);
    gemm_wmma_f32_kernel<<<gM512, blk, 0, stream>>>(xn, Wf1, bf1, nullptr, nullptr, ff, N_, 128, 512, 1);
    gemm_wmma_f32_kernel<<<gM128, blk, 0, stream>>>(ff, Wf2, bf2, out, nullptr, out, N_, 512, 128, 0);
}